// LargeLBCS_29884382446115
// MI455X (gfx1250) — compile-verified
//
#include <hip/hip_runtime.h>
#include <hip/hip_bf16.h>

// LargeLBCS loss on MI455X (gfx1250, wave32).
// Per-q GEMM [16n x 3p] x [3p x 16s], P padded 3->4, on V_WMMA_F32_16X16X4_F32.
// Product over q = elementwise v8f multiplies (identical C/D layout per q).
// A fragments (pauli) live in 40 VGPRs, reused across all 25 s-tiles.
// B fragments come from a pre-padded [s][q][4] heads^2 array in scratch:
// one aligned global_load_b64 per (s-tile, q), fully branch-free.

typedef __attribute__((ext_vector_type(2))) float v2f;
typedef __attribute__((ext_vector_type(4))) float v4f;
typedef __attribute__((ext_vector_type(8))) float v8f;

#define N_WORDS 20000
#define N_S     400
#define N_Q     20
#define N_P     3
#define N_TILES (N_WORDS / 16)          // 1250
#define S_TILES (N_S / 16)              // 25
#define PAD_B_OFF_BYTES 8192            // partials live in d_ws[0..1249]
#define PAD_B_FLOATS    (N_S * N_Q * 4) // 32000 floats = 128 KB

static __device__ __forceinline__ v8f wmma16x16x4(v2f a, v2f b, v8f c) {
    return __builtin_amdgcn_wmma_f32_16x16x4_f32(
        /*neg_a=*/false, a, /*neg_b=*/false, b,
        /*c_mod=*/(short)0, c, /*reuse_a=*/false, /*reuse_b=*/false);
}

// ---------------------------------------------------------------------------
// Kernel 1: squares -> d_out[1..400]=ratios^2, d_out[401..]=heads^2, and
// (optionally) the K=4-padded B operand {h0^2,h1^2,h2^2,0} into scratch.
// ---------------------------------------------------------------------------
__global__ __launch_bounds__(256) void lbcs_prep_kernel(
    const float* __restrict__ heads_param,   // [S,Q,P]
    const float* __restrict__ ratios_param,  // [S]
    float* __restrict__ out,                 // [1+400+24000]
    float* __restrict__ bpad)                // [S*Q*4] or nullptr
{
    int i = blockIdx.x * blockDim.x + threadIdx.x;
    if (i < N_S) {
        float r = ratios_param[i];
        out[1 + i] = r * r;
    }
    if (i < N_S * N_Q * N_P) {
        float h = heads_param[i];
        out[1 + N_S + i] = h * h;
    }
    if (bpad != nullptr && i < N_S * N_Q) {
        const float* h = heads_param + i * N_P;
        float h0 = h[0], h1 = h[1], h2 = h[2];
        v4f v;
        v.x = h0 * h0; v.y = h1 * h1; v.z = h2 * h2; v.w = 0.0f;
        *(v4f*)(bpad + i * 4) = v;      // 16B-aligned store
    }
}

// ---------------------------------------------------------------------------
// Kernel 2: one wave per 16-word n-tile. All address selection is branch-free
// (pointer selects + v_cndmask); EXEC stays all-ones through every WMMA.
// ---------------------------------------------------------------------------
template <bool PAD>
__global__ __launch_bounds__(32) void lbcs_coverage_kernel(
    const float* __restrict__ pauli,    // [N,Q,P]
    const float* __restrict__ coeff,    // [N]
    const float* __restrict__ outbuf,   // d_out (squares already written)
    const float* __restrict__ bpad,     // padded B operand (if PAD)
    float* __restrict__ partials)       // [N_TILES] in d_ws
{
    const float* __restrict__ ratios_sq = outbuf + 1;
    const float* __restrict__ heads_sq  = outbuf + 1 + N_S;

    const int  lane   = threadIdx.x;    // 0..31
    const int  m      = lane & 15;
    const bool hi     = lane >= 16;     // hi half carries K = {2,3}
    const int  n_base = blockIdx.x * 16;

    // --- A fragments: 16x4 f32 per q, K=3 zero-padded, branch-free. ---
    v2f A[N_Q];
    {
        const float* prow = pauli + (size_t)(n_base + m) * (N_Q * N_P);
        const float* pa   = prow + (hi ? 2 : 0);   // K0 source
        const float* pb   = prow + 1;              // K1 source (lo half only)
#pragma unroll
        for (int q = 0; q < N_Q; ++q) {
            float a0 = pa[q * N_P];                // plain loads, no predication
            float a1 = pb[q * N_P];
            A[q].x = a0;
            A[q].y = hi ? 0.0f : a1;               // v_cndmask, K=3 pad
        }
    }

    float cov[8];
#pragma unroll
    for (int j = 0; j < 8; ++j) cov[j] = 0.0f;

    for (int st = 0; st < S_TILES; ++st) {
        const int   s = st * 16 + m;
        const float r = ratios_sq[s];

        // --- B fragments for all 20 qubits of this s-tile. ---
        v2f Bf[N_Q];
        if (PAD) {
            // padded row: 20 x {p0,p1,p2,0}; lo reads +0, hi reads +2. 8B aligned.
            const float* brow = bpad + (size_t)s * (N_Q * 4) + (hi ? 2 : 0);
#pragma unroll
            for (int q = 0; q < N_Q; ++q)
                Bf[q] = *(const v2f*)(brow + q * 4);   // global_load_b64
        } else {
            const float* h0 = heads_sq + (size_t)s * (N_Q * N_P) + (hi ? 2 : 0);
            const float* h1 = heads_sq + (size_t)s * (N_Q * N_P) + 1;
#pragma unroll
            for (int q = 0; q < N_Q; ++q) {
                float bx = h0[q * N_P];
                float by = h1[q * N_P];
                Bf[q].x = bx;
                Bf[q].y = hi ? 0.0f : by;
            }
        }

        // --- 20 WMMAs; product over q as shallow multiply trees. ---
        v8f acc = {1.f, 1.f, 1.f, 1.f, 1.f, 1.f, 1.f, 1.f};
        const v8f cz = {0.f, 0.f, 0.f, 0.f, 0.f, 0.f, 0.f, 0.f};
#pragma unroll
        for (int g = 0; g < N_Q; g += 4) {
            v8f c0 = wmma16x16x4(A[g + 0], Bf[g + 0], cz);
            v8f c1 = wmma16x16x4(A[g + 1], Bf[g + 1], cz);
            v8f c2 = wmma16x16x4(A[g + 2], Bf[g + 2], cz);
            v8f c3 = wmma16x16x4(A[g + 3], Bf[g + 3], cz);
            acc *= (c0 * c1) * (c2 * c3);
        }
#pragma unroll
        for (int j = 0; j < 8; ++j) cov[j] += r * acc[j];
    }

    // --- Finish coverage rows: sum the 16 s-lanes in each half. ---
    float part = 0.0f;
#pragma unroll
    for (int j = 0; j < 8; ++j) {
        float v = cov[j];
        v += __shfl_xor(v, 1, 32);
        v += __shfl_xor(v, 2, 32);
        v += __shfl_xor(v, 4, 32);
        v += __shfl_xor(v, 8, 32);        // stays within each 16-lane half
        if (m == 0) {                     // lane 0: row j ; lane 16: row 8+j
            const int   nrow = n_base + (hi ? 8 + j : j);
            const float c    = coeff[nrow];
            part += (c * c) / v;
        }
    }
    part += __shfl_xor(part, 16, 32);     // fold lane16 into lane0
    if (lane == 0) partials[blockIdx.x] = part;
}

// ---------------------------------------------------------------------------
// Kernel 3: deterministic single-block tree reduction of tile partials.
// ---------------------------------------------------------------------------
__global__ __launch_bounds__(256) void lbcs_reduce_kernel(
    const float* __restrict__ partials, float* __restrict__ out, int count)
{
    __shared__ float sm[256];
    float s = 0.0f;
    for (int i = threadIdx.x; i < count; i += 256) s += partials[i];
    sm[threadIdx.x] = s;
    __syncthreads();
#pragma unroll
    for (int off = 128; off > 0; off >>= 1) {
        if ((int)threadIdx.x < off) sm[threadIdx.x] += sm[threadIdx.x + off];
        __syncthreads();
    }
    if (threadIdx.x == 0) out[0] = sm[0];
}

// ---------------------------------------------------------------------------
extern "C" void kernel_launch(void* const* d_in, const int* in_sizes, int n_in,
                              void* d_out, int out_size, void* d_ws, size_t ws_size,
                              hipStream_t stream) {
    const float* heads_param  = (const float*)d_in[0];  // [S,Q,P]
    const float* ratios_param = (const float*)d_in[1];  // [S]
    const float* pauli        = (const float*)d_in[2];  // [N,Q,P]
    const float* coeff        = (const float*)d_in[3];  // [N]
    float* out      = (float*)d_out;
    float* partials = (float*)d_ws;

    const size_t need    = (size_t)PAD_B_OFF_BYTES + (size_t)PAD_B_FLOATS * sizeof(float);
    const bool   use_pad = ws_size >= need;
    float* bpad = use_pad ? (float*)((char*)d_ws + PAD_B_OFF_BYTES) : nullptr;

    const int prep_elems = N_S * N_Q * N_P;              // 24000 dominates
    lbcs_prep_kernel<<<(prep_elems + 255) / 256, 256, 0, stream>>>(
        heads_param, ratios_param, out, bpad);

    if (use_pad)
        lbcs_coverage_kernel<true><<<N_TILES, 32, 0, stream>>>(
            pauli, coeff, out, bpad, partials);
    else
        lbcs_coverage_kernel<false><<<N_TILES, 32, 0, stream>>>(
            pauli, coeff, out, nullptr, partials);

    lbcs_reduce_kernel<<<1, 256, 0, stream>>>(partials, out, N_TILES);
}